// SeesawLossWithLogits_5076651344113
// MI455X (gfx1250) — compile-verified
//
#include <hip/hip_runtime.h>
#include <hip/hip_bf16.h>
#include <stdint.h>

#define CCLS   1230          // class count C
#define C2     615           // float2 elements per row (1230/2, exact)
#define ROWS   8             // one wave32 per row, 8 waves per block
#define K2     20            // ceil(615/32) strided float2 iterations per lane
#define EPSV   1e-6f

typedef float v2f __attribute__((ext_vector_type(2)));

// ---------------------------------------------------------------------------
// Seesaw loss, one-hot reduction (denom GEMM collapsed to a gathered dot):
//   loss[n] = log( dot(s[l,:], exp(z - m)) + eps ) - (z[l] - m),  l = label[n]
//
// Kernel 1: one wave32 per sample row.
//   Phase A: async-stage logits row -> LDS  (global_load_async_to_lds_b64,
//            th:TH_LOAD_NT — logits are single-use, keep them out of L2)
//   Phase B: scan targets row (non-temporal) -> find one-hot label l
//   Phase C: s_wait_asynccnt 0
//   Phase D: row max from LDS (wave reduce)
//   Phase E: acc = sum_j s[l,j] * exp(z_j - m)   (s row gather stays L2-hot)
//   Phase F: per-row loss; block writes one partial sum of its 8 rows.
// ---------------------------------------------------------------------------
__global__ __launch_bounds__(256) void seesaw_rows_kernel(
    const float* __restrict__ logits,
    const float* __restrict__ targets,
    const float* __restrict__ smat,
    float* __restrict__ partial,
    int n_rows)
{
    // 1232 floats per row keeps each LDS row 16B-aligned (4928 B stride)
    __shared__ float lds[ROWS][CCLS + 2];
    __shared__ float lsum[ROWS];

    const int lane = threadIdx.x & 31;
    const int w    = threadIdx.x >> 5;
    const int n    = blockIdx.x * ROWS + w;
    if (blockIdx.x * ROWS >= n_rows) return;   // uniform per block

    const float* grow = logits  + (size_t)n * CCLS;
    const v2f*   trow = (const v2f*)(targets + (size_t)n * CCLS);

    // wave-uniform LDS byte offset of this wave's row (low 32 bits of flat ptr)
    const uint32_t ldsRow = (uint32_t)(uintptr_t)(&lds[w][0]);

    // ---- Phase A: async logits row -> LDS (CDNA5 async engine, ASYNCcnt) ----
    #pragma unroll
    for (int k = 0; k < K2; ++k) {
        int i2 = k * 32 + lane;
        if (i2 < C2) {
            uint32_t ldsoff = ldsRow + (uint32_t)(i2 * 8);
            const float* gp = grow + i2 * 2;           // 8B-aligned (4920 % 8 == 0)
            asm volatile("global_load_async_to_lds_b64 %0, %1, off th:TH_LOAD_NT"
                         :: "v"(ldsoff), "v"(gp)
                         : "memory");
        }
    }

    // ---- Phase B: find one-hot label while async loads are in flight ----
    int myl = -1;
    #pragma unroll
    for (int k = 0; k < K2; ++k) {
        int i2 = k * 32 + lane;
        if (i2 < C2) {
            v2f t = __builtin_nontemporal_load(trow + i2);  // single-use stream
            if (t.x > 0.5f) myl = i2 * 2;
            if (t.y > 0.5f) myl = i2 * 2 + 1;
        }
    }
    #pragma unroll
    for (int off = 16; off > 0; off >>= 1)
        myl = max(myl, __shfl_xor(myl, off, 32));
    const int l = myl;

    // warm the gathered s row (global_prefetch_b8); s is L2-resident (6 MB)
    const float* srow = smat + (size_t)l * CCLS;
    __builtin_prefetch(srow + lane * 39, 0, 0);

    // ---- Phase C: wait for the async engine to land the logits row ----
    asm volatile("s_wait_asynccnt 0" ::: "memory");

    // ---- Phase D: numerically-stable row max ----
    float m = -INFINITY;
    #pragma unroll
    for (int k = 0; k < K2; ++k) {
        int i2 = k * 32 + lane;
        if (i2 < C2) {
            float2 v = *(const float2*)&lds[w][i2 * 2];
            m = fmaxf(m, fmaxf(v.x, v.y));
        }
    }
    #pragma unroll
    for (int off = 16; off > 0; off >>= 1)
        m = fmaxf(m, __shfl_xor(m, off, 32));

    // ---- Phase E: acc = dot(s[l,:], exp(row - m)) ----
    float acc = 0.0f;
    #pragma unroll
    for (int k = 0; k < K2; ++k) {
        int i2 = k * 32 + lane;
        if (i2 < C2) {
            float2 v  = *(const float2*)&lds[w][i2 * 2];
            float2 sv = *(const float2*)(srow + i2 * 2);   // 8B-aligned: l*4920 % 8 == 0
            acc += sv.x * __expf(v.x - m) + sv.y * __expf(v.y - m);
        }
    }
    #pragma unroll
    for (int off = 16; off > 0; off >>= 1)
        acc += __shfl_xor(acc, off, 32);

    // ---- Phase F: per-row loss ----
    if (lane == 0) {
        float zl    = lds[w][l];
        float el    = __expf(zl - m);
        float sll   = srow[l];                 // == 1.0f by construction; robust anyway
        float denom = acc + el - sll * el;     // = sum_{j!=l} s[l,j] e_j + e_l
        lsum[w] = __logf(denom + EPSV) - (zl - m);
    }
    __syncthreads();
    if (threadIdx.x == 0) {
        float t = 0.0f;
        #pragma unroll
        for (int i = 0; i < ROWS; ++i) t += lsum[i];
        partial[blockIdx.x] = t;
    }
}

// ---------------------------------------------------------------------------
// Kernel 2: deterministic single-block finish: mean over all rows.
// ---------------------------------------------------------------------------
__global__ __launch_bounds__(256) void seesaw_finish_kernel(
    const float* __restrict__ partial, float* __restrict__ out,
    int nblocks, float inv_n)
{
    __shared__ float sm[256];
    float t = 0.0f;
    for (int i = threadIdx.x; i < nblocks; i += 256) t += partial[i];
    sm[threadIdx.x] = t;
    __syncthreads();
    #pragma unroll
    for (int s2 = 128; s2 > 0; s2 >>= 1) {
        if ((int)threadIdx.x < s2) sm[threadIdx.x] += sm[threadIdx.x + s2];
        __syncthreads();
    }
    if (threadIdx.x == 0) out[0] = sm[0] * inv_n;
}

extern "C" void kernel_launch(void* const* d_in, const int* in_sizes, int n_in,
                              void* d_out, int out_size, void* d_ws, size_t ws_size,
                              hipStream_t stream) {
    const float* logits  = (const float*)d_in[0];
    const float* targets = (const float*)d_in[1];
    const float* smat    = (const float*)d_in[2];
    float* partial = (float*)d_ws;

    const int n_rows  = in_sizes[0] / CCLS;           // 32768
    const int nblocks = (n_rows + ROWS - 1) / ROWS;   // 4096

    seesaw_rows_kernel<<<nblocks, 256, 0, stream>>>(logits, targets, smat,
                                                    partial, n_rows);
    seesaw_finish_kernel<<<1, 256, 0, stream>>>(partial, (float*)d_out,
                                                nblocks, 1.0f / (float)n_rows);
}